// Net_77867757076852
// MI455X (gfx1250) — compile-verified
//
#include <hip/hip_runtime.h>

typedef __attribute__((ext_vector_type(2))) float v2f;
typedef __attribute__((ext_vector_type(8))) float v8f;

#define DT_F 0.1f

struct alignas(4) F3 { float a, b, c; };   // 12B packed row for b96-able stores

// SWAPX16 via ds_swizzle (group-of-32: xor=0x10, or=0, and=0x1f) -> lane L <-> L^16
__device__ __forceinline__ float swap16(float v) {
  return __int_as_float(__builtin_amdgcn_ds_swizzle(__float_as_int(v), 0x401F));
}

__device__ __forceinline__ void lorenz(float x, float y, float z,
                                       float sg, float rh, float bt,
                                       float& d0, float& d1, float& d2) {
  d0 = sg * (y - x);
  d1 = rh * x - y - x * z;
  d2 = x * y - bt * z;
}

__global__ __launch_bounds__(256) void mlp_lorenz_rk4(
    const float* __restrict__ x,
    const float* __restrict__ W1, const float* __restrict__ b1,
    const float* __restrict__ W2, const float* __restrict__ b2,
    const float* __restrict__ sigma_p, const float* __restrict__ rho_p,
    const float* __restrict__ beta_p,
    float* __restrict__ out, int numTiles) {
  const int lane = threadIdx.x & 31;
  const int m    = lane & 15;
  const bool hi  = lane >= 16;
  const int waveId = (blockIdx.x * blockDim.x + threadIdx.x) >> 5;
  const int nWaves = (gridDim.x * blockDim.x) >> 5;

  const float sg = sigma_p[0], rh = rho_p[0], bt = beta_p[0];
  const float b2_0 = b2[0], b2_1 = b2[1], b2_2 = b2[2];

  // ---- wave-invariant weight tiles ----
  // Layer 1 A-matrix (16x4 f32): M=hidden(0..9 valid), K=input(0..2 valid).
  // A layout: lanes 0-15 -> {v0=K0, v1=K1}; lanes 16-31 -> {v0=K2, v1=K3}.
  v2f a1;
  a1.x = (m < 10) ? W1[m * 3 + (hi ? 2 : 0)] : 0.0f;
  a1.y = (m < 10 && !hi) ? W1[m * 3 + 1] : 0.0f;

  // Layer 2 A-matrices, K=hidden split 4+4+2: M=out-dim(0..2 valid).
  v2f a2a, a2b, a2c;
  a2a.x = (m < 3) ? W2[m * 10 + (hi ? 2 : 0)] : 0.0f;
  a2a.y = (m < 3) ? W2[m * 10 + (hi ? 3 : 1)] : 0.0f;
  a2b.x = (m < 3) ? W2[m * 10 + (hi ? 6 : 4)] : 0.0f;
  a2b.y = (m < 3) ? W2[m * 10 + (hi ? 7 : 5)] : 0.0f;
  a2c.x = (m < 3 && !hi) ? W2[m * 10 + 8] : 0.0f;
  a2c.y = (m < 3 && !hi) ? W2[m * 10 + 9] : 0.0f;

  // b1 broadcast into D-layout slots: VGPR v holds M=v (lo lanes) / M=v+8 (hi lanes).
  float bb[8];
#pragma unroll
  for (int v = 0; v < 8; ++v) {
    const int idx = hi ? (v + 8) : v;
    bb[v] = (idx < 10) ? b1[idx] : 0.0f;
  }

  const v8f czero = {};

  for (int t = waveId; t < numTiles; t += nWaves) {
    const int rb3 = (t * 16 + m) * 3;   // row-major [B,3] base for this lane's row

    // X^T as B-matrix (4x16): v0 = {lo:K0, hi:K2}, v1 = {lo:K1, hi:K3(pad 0)}.
    // Both loads are unconditional (rb3+1 < 3B for every lane) -> no exec branching.
    const float bxx = x[rb3 + (hi ? 2 : 0)];
    const float bxy = x[rb3 + 1];
    v2f bx;
    bx.x = bxx;
    bx.y = hi ? 0.0f : bxy;

    // Prefetch next grid-stride tile (clamped so the address stays valid).
    {
      int tn = t + nWaves;
      if (tn >= numTiles) tn = t;
      __builtin_prefetch(x + (tn * 16 + m) * 3, 0, 1);
    }

    // Layer 1: H[16h x 16n] = W1 x X^T
    v8f h = __builtin_amdgcn_wmma_f32_16x16x4_f32(false, a1, false, bx,
                                                  (short)0, czero, false, false);
    // bias + ReLU (hidden index is the VGPR slot -> pure per-slot op)
#pragma unroll
    for (int v = 0; v < 8; ++v) h[v] = fmaxf(h[v] + bb[v], 0.0f);

    // D-layout -> B-layout for layer 2 K-chunks: lane-half swaps (full-EXEC, convergent)
    const float s0 = swap16(h[0]);
    const float s1 = swap16(h[1]);
    const float s2 = swap16(h[2]);
    const float s3 = swap16(h[3]);
    const float s6 = swap16(h[6]);
    const float s7 = swap16(h[7]);

    v2f bc0, bc1, bc2;
    bc0.x = hi ? s2 : h[0];  // K=0 (lo) / K=2 (hi)
    bc0.y = hi ? s3 : h[1];  // K=1 / K=3
    bc1.x = hi ? s6 : h[4];  // K=4 / K=6
    bc1.y = hi ? s7 : h[5];  // K=5 / K=7
    bc2.x = hi ? 0.0f : s0;  // K=8 / K=10(pad)
    bc2.y = hi ? 0.0f : s1;  // K=9 / K=11(pad)

    // Layer 2: R[3 x 16n] = W2 x H, accumulated over three K=4 chunks
    v8f d2 = __builtin_amdgcn_wmma_f32_16x16x4_f32(false, a2a, false, bc0,
                                                   (short)0, czero, false, false);
    d2 = __builtin_amdgcn_wmma_f32_16x16x4_f32(false, a2b, false, bc1,
                                               (short)0, d2, false, false);
    d2 = __builtin_amdgcn_wmma_f32_16x16x4_f32(false, a2c, false, bc2,
                                               (short)0, d2, false, false);

    if (!hi) {  // lo half holds all 3 state components of its row; hi half skips RK4+store
      const float r0 = d2[0] + b2_0;
      const float r1 = d2[1] + b2_1;
      const float r2 = d2[2] + b2_2;

      float k10, k11, k12, k20, k21, k22, k30, k31, k32, k40, k41, k42;
      lorenz(r0, r1, r2, sg, rh, bt, k10, k11, k12);
      lorenz(r0 + 0.5f * DT_F * k10, r1 + 0.5f * DT_F * k11, r2 + 0.5f * DT_F * k12,
             sg, rh, bt, k20, k21, k22);
      lorenz(r0 + 0.5f * DT_F * k20, r1 + 0.5f * DT_F * k21, r2 + 0.5f * DT_F * k22,
             sg, rh, bt, k30, k31, k32);
      lorenz(r0 + DT_F * k30, r1 + DT_F * k31, r2 + DT_F * k32,
             sg, rh, bt, k40, k41, k42);
      const float c = DT_F / 6.0f;

      F3 o;
      o.a = r0 + c * (k10 + 2.0f * k20 + 2.0f * k30 + k40);
      o.b = r1 + c * (k11 + 2.0f * k21 + 2.0f * k31 + k41);
      o.c = r2 + c * (k12 + 2.0f * k22 + 2.0f * k32 + k42);
      reinterpret_cast<F3*>(out)[t * 16 + m] = o;  // 12B packed -> b96-able store
    }
  }
}

extern "C" void kernel_launch(void* const* d_in, const int* in_sizes, int n_in,
                              void* d_out, int out_size, void* d_ws, size_t ws_size,
                              hipStream_t stream) {
  const float* x     = (const float*)d_in[0];
  const float* W1    = (const float*)d_in[1];
  const float* b1    = (const float*)d_in[2];
  const float* W2    = (const float*)d_in[3];
  const float* b2    = (const float*)d_in[4];
  const float* sigma = (const float*)d_in[5];
  const float* rho   = (const float*)d_in[6];
  const float* beta  = (const float*)d_in[7];
  float* out = (float*)d_out;

  const int Bn = in_sizes[0] / 3;      // 8388608
  const int numTiles = Bn / 16;        // B divisible by 16

  const int block = 256;               // 8 waves per block
  int grid = 8192;                     // 65536 waves, grid-stride over tiles
  const int maxBlocks = (numTiles * 32 + block - 1) / block;
  if (grid > maxBlocks) grid = maxBlocks > 0 ? maxBlocks : 1;

  mlp_lorenz_rk4<<<grid, block, 0, stream>>>(x, W1, b1, W2, b2, sigma, rho, beta,
                                             out, numTiles);
}